// STPPReorganized_46231027974782
// MI455X (gfx1250) — compile-verified
//
#include <hip/hip_runtime.h>
#include <hip/hip_bf16.h>

// ---- problem constants (from reference) ----
#define TROWS   10000          // T_TOTAL
#define NCOL    4005           // FEAT_DIM = (201+200+400)*5
#define NPROPS  2000
#define CH      8              // rows per chunk
#define NCHUNK  1250           // TROWS / CH (exact)
#define NTILE   251            // ceil(NCOL/16) column tiles for WMMA pass
#define CPS     25             // chunks per super-chunk (200 rows)
#define NSUPER  50             // NCHUNK / CPS (exact)

typedef float v2f __attribute__((ext_vector_type(2)));
typedef float v8f __attribute__((ext_vector_type(8)));

__device__ __forceinline__ int imin(int a, int b) { return a < b ? a : b; }
__device__ __forceinline__ int imax(int a, int b) { return a > b ? a : b; }

// ------------------------------------------------------------------
// Pass 1: one wave per (16-col tile, super-chunk of 200 rows).
// Running column sums via a single chained V_WMMA_F32_16X16X4_F32
// accumulator (all-ones A => every row of D = column sums; result is
// invariant to the K<->(half,vgpr) layout since A is all ones).
// Before folding chunk `ch` in, acc[0] (= super-local exclusive
// prefix) is stored to Pexcl[ch]; after 25 chunks the super total is
// stored to S[super]. EXEC is all-ones at every WMMA: the grid is
// exact, edge columns are handled branchlessly, and the store guards
// are structured ifs that re-converge before the next WMMA.
// ------------------------------------------------------------------
__global__ void colsum_prefix_kernel(const float* __restrict__ x,
                                     float* __restrict__ Pexcl,
                                     float* __restrict__ S) {
  const int wave = blockIdx.x * 2 + (threadIdx.x >> 5);     // 2 waves / block
  const int tile = wave % NTILE;
  const int sup  = wave / NTILE;                            // < NSUPER exactly
  const int lane = threadIdx.x & 31;
  const int half = lane >> 4;                               // 0 or 1
  const int cl   = lane & 15;
  const int col  = tile * 16 + cl;

  const bool  colOK    = (col < NCOL);
  const int   colC     = colOK ? col : (NCOL - 1);
  const float msk      = colOK ? 1.0f : 0.0f;               // v_cndmask, no EXEC change
  const bool  isWriter = (half == 0) && colOK;

  v2f a; a.x = 1.0f; a.y = 1.0f;                            // all-ones A (16x4 f32)
  v8f acc = {0.f, 0.f, 0.f, 0.f, 0.f, 0.f, 0.f, 0.f};

  int ch = sup * CPS;
  const float* xrow = x + (size_t)ch * CH * NCOL + colC;

#pragma unroll 5
  for (int cc = 0; cc < CPS; ++cc, ++ch) {
    // super-local EXCLUSIVE prefix for this chunk (store latches data;
    // ISA: store followed by overwrite of source GPRs is hazard-free)
    if (isWriter) Pexcl[(size_t)ch * NCOL + col] = acc[0];

#pragma unroll
    for (int kk = 0; kk < 2; ++kk) {                        // 2 WMMAs x K=4 = 8 rows
      v2f b;
      b.x = xrow[(size_t)(kk * 4 + half)     * NCOL] * msk; // rows +0/+1
      b.y = xrow[(size_t)(kk * 4 + 2 + half) * NCOL] * msk; // rows +2/+3
      acc = __builtin_amdgcn_wmma_f32_16x16x4_f32(
          /*neg_a=*/false, a, /*neg_b=*/false, b,
          /*c_mod=*/(short)0, acc, /*reuse_a=*/false, /*reuse_b=*/false);
    }
    xrow += (size_t)CH * NCOL;
  }

  if (isWriter) S[(size_t)sup * NCOL + col] = acc[0];       // super total
}

// ------------------------------------------------------------------
// Pass 2: in-place exclusive scan of the 50 super totals per column.
// Fully unrolled: 50 batched coalesced loads, one wait, 51 stores.
// ------------------------------------------------------------------
__global__ void scan_supers_kernel(float* __restrict__ S) {
  const int col = blockIdx.x * blockDim.x + threadIdx.x;
  if (col >= NCOL) return;
  float run = 0.0f;
#pragma unroll
  for (int sp = 0; sp < NSUPER; ++sp) {
    const size_t idx = (size_t)sp * NCOL + col;
    const float t = S[idx];
    S[idx] = run;
    run += t;
  }
  S[(size_t)NSUPER * NCOL + col] = run;                     // csum at row 10000
}

// csum[r][col] (sum of x rows [0, r)) = super prefix + chunk-local
// exclusive prefix + tail of <= 7 rows of x (L2-resident after pass 1).
// All branches are wave-uniform (r depends only on the proposal).
__device__ __forceinline__ float csum_at(const float* __restrict__ x,
                                         const float* __restrict__ Pexcl,
                                         const float* __restrict__ S,
                                         int r, int col) {
  const int ch  = r >> 3;                 // [0, 1250]
  const int rem = r & 7;
  const int sup = ch / CPS;               // [0, 50]
  if (sup >= NSUPER) {                    // r == 10000 (rem == 0)
    return S[(size_t)NSUPER * NCOL + col];
  }
  float v = S[(size_t)sup * NCOL + col] + Pexcl[(size_t)ch * NCOL + col];
  const float* xp = x + (size_t)(ch << 3) * NCOL + col;
  for (int q = 0; q < rem; ++q) v += xp[(size_t)q * NCOL];
  return v;
}

// PARTS = [(0,1,0),(1,1,0),(1,2,0),(1,2,1),(2,1,0)]
__constant__ int   c_part_s[5]  = {0, 1, 1, 1, 2};
__constant__ float c_part_np[5] = {1.f, 1.f, 2.f, 2.f, 1.f};
__constant__ float c_part_i[5]  = {0.f, 0.f, 0.f, 1.f, 0.f};

// ------------------------------------------------------------------
// Pass 3: pooled segment means. One thread per (proposal, score col).
// blockIdx.y = proposal -> ticks/scaling loads scalarize; x/Pexcl/S
// reads are coalesced across c within each part segment.
// ------------------------------------------------------------------
__global__ void pspool_kernel(const float* __restrict__ x,
                              const int*   __restrict__ ticks,
                              const float* __restrict__ scal,
                              const float* __restrict__ Pexcl,
                              const float* __restrict__ S,
                              float* __restrict__ out,
                              int L, int colBase, int outBase) {
  const int p = blockIdx.y;
  const int c = blockIdx.x * blockDim.x + threadIdx.x;
  if (c >= L) return;

  float tk[4];
  tk[0] = (float)ticks[p * 4 + 0];
  tk[1] = (float)ticks[p * 4 + 1];
  tk[2] = (float)ticks[p * 4 + 2];
  tk[3] = (float)ticks[p * 4 + 3];
  const float s0 = scal[p * 2 + 0];
  const float s1 = scal[p * 2 + 1];

  float acc = 0.0f;
#pragma unroll
  for (int off = 0; off < 5; ++off) {
    const int   s_idx = c_part_s[off];
    const float npart = c_part_np[off];
    const float fi    = c_part_i[off];

    const float left  = tk[s_idx];
    const float right = fmaxf(left + 1.0f, tk[s_idx + 1]);
    const bool  skip  = (right <= 0.0f) || (left >= (float)TROWS);
    const float s     = (s_idx == 0) ? s0 : ((s_idx == 2) ? s1 : 1.0f);

    const float step = (right - left) / npart;
    const int pl = (int)(left + fi * step);          // trunc toward 0 == astype(int32)
    const int pr = (int)(left + (fi + 1.0f) * step);
    const bool valid = ((pr - pl) >= 1) && !skip;
    const int plc = imin(imax(pl, 0), TROWS);
    const int prc = imin(imax(pr, 0), TROWS);

    const int col = colBase + off * L + c;
    const float seg   = csum_at(x, Pexcl, S, prc, col)
                      - csum_at(x, Pexcl, S, plc, col);
    const float denom = (float)imax(pr - pl, 1);
    acc += valid ? ((seg / denom) * s) : 0.0f;
  }

  out[(size_t)outBase + (size_t)p * L + c] = acc;
}

extern "C" void kernel_launch(void* const* d_in, const int* in_sizes, int n_in,
                              void* d_out, int out_size, void* d_ws, size_t ws_size,
                              hipStream_t stream) {
  (void)in_sizes; (void)n_in; (void)out_size; (void)ws_size;

  const float* x     = (const float*)d_in[0];   // (10000, 4005) f32
  const int*   ticks = (const int*)  d_in[1];   // (2000, 4) i32
  const float* scal  = (const float*)d_in[2];   // (2000, 2) f32
  float*       out   = (float*)d_out;           // act|comp|reg concat: 2000*801 f32

  // Workspace: Pexcl (1250 x 4005 f32, ~20 MB) + S (51 x 4005 f32, ~0.8 MB).
  float* Pexcl = (float*)d_ws;
  float* S     = Pexcl + (size_t)NCHUNK * NCOL;

  // Pass 1: WMMA-chained column sums + super-local exclusive prefixes.
  {
    const int totalWaves = NTILE * NSUPER;      // 251 * 50 = 12550
    colsum_prefix_kernel<<<dim3(totalWaves / 2), dim3(64), 0, stream>>>(x, Pexcl, S);
  }

  // Pass 2: tiny in-place exclusive scan over 50 super totals per column.
  scan_supers_kernel<<<dim3((NCOL + 255) / 256), dim3(256), 0, stream>>>(S);

  // Pass 3: pooled means per group (act L=201, comp L=200, reg L=400).
  {
    dim3 blk(128);
    pspool_kernel<<<dim3((201 + 127) / 128, NPROPS), blk, 0, stream>>>(
        x, ticks, scal, Pexcl, S, out, 201, 0,    0);
    pspool_kernel<<<dim3((200 + 127) / 128, NPROPS), blk, 0, stream>>>(
        x, ticks, scal, Pexcl, S, out, 200, 1005, NPROPS * 201);
    pspool_kernel<<<dim3((400 + 127) / 128, NPROPS), blk, 0, stream>>>(
        x, ticks, scal, Pexcl, S, out, 400, 2005, NPROPS * 201 + NPROPS * 200);
  }
}